// BagOfWordsClassifier_5420248727899
// MI455X (gfx1250) — compile-verified
//
#include <hip/hip_runtime.h>
#include <hip/hip_bf16.h>
#include <stdint.h>

// BagOfWordsClassifier on MI455X (gfx1250).
//
// logits[b,c] = sum_s (ids[b,s]!=0) * W[c, ids[b,s]] + bias[c]
//   B=512, S=512, V=50257, C=2.
//
// Roofline: ~1 MFLOP, ~1 MB of ids + 512K random 4B gathers into a 402 KB
// table that stays L2-resident (192 MB L2). Pure gather-latency-bound;
// no dense matmul exists, so WMMA is intentionally absent. The plan:
//  * 1 block per batch row, 256 threads (8 wave32s) -> 4096 waves in
//    flight to hide random-gather latency.
//  * Row of 512 int32 ids (2 KB) is DMA'd global->LDS with the CDNA5
//    Tensor Data Mover (tensor_load_to_lds), issued by wave 0 only
//    (TDM ignores EXEC; the uniform t<32 branch becomes s_cbranch_execz,
//    so other waves never issue it), completion via s_wait_tensorcnt 0
//    + workgroup barrier. Frees the vector-memory path for the W gathers.
//  * Each thread handles 2 tokens, gathers W[0][id], W[1][id]
//    unconditionally (mask-multiply, keeps all 4 loads in flight).
//  * Reduction: wave32 shfl_xor tree, then per-wave partials to LDS and a
//    fixed-order sum by thread 0 (deterministic; no float atomics).

#define VOCAB 50257
#define SEQ   512
#define NWAVES 8   // 256 threads / wave32

typedef unsigned int u32;
typedef __attribute__((ext_vector_type(4))) u32 v4u_t;
typedef __attribute__((ext_vector_type(8))) int v8i_t;
typedef __attribute__((ext_vector_type(4))) int v4i_t;

__global__ __launch_bounds__(256) void bow_classifier_kernel(
    const int*   __restrict__ ids,   // [B, SEQ]
    const float* __restrict__ W,     // [2, VOCAB] row-major
    const float* __restrict__ bias,  // [2]
    float*       __restrict__ out)   // [B, 2]
{
    __shared__ __align__(16) int s_ids[SEQ];
    __shared__ float s_part[NWAVES][2];

    const int row = blockIdx.x;
    const int t   = (int)threadIdx.x;

#if __has_builtin(__builtin_amdgcn_tensor_load_to_lds)
    // ---- TDM: DMA this row's 512 ids (2 KB) into LDS -------------------
    if (t < 32) {  // wave 0 only (uniform branch -> s_cbranch_execz skips)
        const unsigned long long gaddr =
            (unsigned long long)(uintptr_t)(ids + (size_t)row * SEQ);
        // Flat shared pointer: addr[31:0] is the LDS byte offset (aperture
        // layout, ISA 10.2).
        const u32 lds_off = (u32)(uintptr_t)(&s_ids[0]);

        // D# group 0 (CDNA5 ISA):
        //   [1:0] count=1, [2] is_restore=0, [30] gather_idx_size=0,
        //   [31] gather_mode=0, [63:32] lds_addr, [120:64] global_addr,
        //   [127:126] type=2 ("image").
        v4u_t g0;
        g0.x = 1u;
        g0.y = lds_off;
        g0.z = (u32)(gaddr & 0xFFFFFFFFull);
        g0.w = (u32)((gaddr >> 32) & 0x01FFFFFFull) | (2u << 30);

        // D# group 1 (§8.4): workgroup_mask=0 (not in cluster),
        //   data_size=2 (4 B), no barrier/iterate/pad,
        //   tensor_dim0=512, tensor_dim1=1, tile_dim0=512, tile_dim1=1,
        //   tensor_dim0_stride=512.
        v8i_t g1;
        g1[0] = (int)(2u << 16);        // data_size=4B
        g1[1] = (int)((u32)SEQ << 16);  // tensor_dim0[15:0] @ bits 63:48
        g1[2] = (int)(1u << 16);        // tensor_dim1[15:0] @ bits 95:80
        g1[3] = (int)((u32)SEQ << 16);  // tile_dim0 @ bits 127:112
        g1[4] = 1;                      // tile_dim1=1, tile_dim2=0
        g1[5] = SEQ;                    // tensor_dim0_stride[31:0]
        g1[6] = 0;
        g1[7] = 0;

        v4i_t g2 = {0, 0, 0, 0};        // dims 2/3 unused (2D mode)
        v4i_t g3 = {0, 0, 0, 0};

#if __has_include(<hip/amd_detail/amd_gfx1250_TDM.h>)
        // amdgpu-toolchain (clang-23): 6-arg form
        v8i_t gx = {0, 0, 0, 0, 0, 0, 0, 0};
        __builtin_amdgcn_tensor_load_to_lds(g0, g1, g2, g3, gx, 0);
#else
        // ROCm 7.2 (clang-22): 5-arg form
        __builtin_amdgcn_tensor_load_to_lds(g0, g1, g2, g3, 0);
#endif
        __builtin_amdgcn_s_wait_tensorcnt(0);
    }
#else
    // Fallback: cooperative LDS fill (2 ids per thread, coalesced).
    {
        const int* rp = ids + (size_t)row * SEQ;
        s_ids[t]       = rp[t];
        s_ids[t + 256] = rp[t + 256];
    }
#endif
    __syncthreads();

    // ---- Gather-accumulate: 2 tokens per thread ------------------------
    const int2 tok = ((const int2*)s_ids)[t];  // tokens 2t, 2t+1

    const float m0 = (tok.x != 0) ? 1.0f : 0.0f;
    const float m1 = (tok.y != 0) ? 1.0f : 0.0f;

    // Unconditional gathers (id 0 is a valid index), masked by multiply so
    // all four loads issue back-to-back and overlap in the memory system.
    const float w00 = W[tok.x];
    const float w01 = W[(size_t)VOCAB + tok.x];
    const float w10 = W[tok.y];
    const float w11 = W[(size_t)VOCAB + tok.y];

    float a0 = fmaf(m0, w00, m1 * w10);   // class 0 partial
    float a1 = fmaf(m0, w01, m1 * w11);   // class 1 partial

    // ---- wave32 reduction (deterministic butterfly) --------------------
    #pragma unroll
    for (int off = 16; off > 0; off >>= 1) {
        a0 += __shfl_xor(a0, off, 32);
        a1 += __shfl_xor(a1, off, 32);
    }

    // ---- cross-wave reduction: fixed-order, no float atomics -----------
    const int wave = t >> 5;
    if ((t & 31) == 0) {
        s_part[wave][0] = a0;
        s_part[wave][1] = a1;
    }
    __syncthreads();

    if (t == 0) {
        float r0 = s_part[0][0];
        float r1 = s_part[0][1];
        #pragma unroll
        for (int w = 1; w < NWAVES; ++w) {  // fixed order -> bit-reproducible
            r0 += s_part[w][0];
            r1 += s_part[w][1];
        }
        out[(size_t)row * 2 + 0] = r0 + bias[0];
        out[(size_t)row * 2 + 1] = r1 + bias[1];
    }
}

extern "C" void kernel_launch(void* const* d_in, const int* in_sizes, int n_in,
                              void* d_out, int out_size, void* d_ws, size_t ws_size,
                              hipStream_t stream) {
    const int*   ids  = (const int*)d_in[0];    // [B, SEQ] int32
    const float* W    = (const float*)d_in[1];  // [2, VOCAB] f32
    const float* bias = (const float*)d_in[2];  // [2] f32
    float*       out  = (float*)d_out;          // [B, 2] f32

    const int rows = in_sizes[0] / SEQ;         // B = 512
    bow_classifier_kernel<<<dim3(rows), dim3(256), 0, stream>>>(ids, W, bias, out);
}